// LISTASparseEncoder_83726092469086
// MI455X (gfx1250) — compile-verified
//
#include <hip/hip_runtime.h>
#include <math.h>

typedef __attribute__((ext_vector_type(2))) float v2f;
typedef __attribute__((ext_vector_type(8))) float v8f;

#define FEAT 512          // feature_dim == C
#define SDIM 1024         // sparse_dim
#define HW   1024         // H*W
#define NBATCH 32
#define NT   32           // columns (spatial positions) per tile / workgroup
#define ZSTRIDE 40        // LDS row stride: 2*40 % 64 == 16 -> conflict-free B loads
#define NWAVES 16
#define ROWS_PER_WAVE (SDIM / NWAVES)   // 64

static __device__ inline v8f vzero8() {
    v8f v;
#pragma unroll
    for (int i = 0; i < 8; ++i) v[i] = 0.0f;
    return v;
}

// Fused LISTA: Wx GEMM + 5 iterations of (z+=Wx; z-=S@z; soft-thresh) + transposed writeout.
// One workgroup = 32 columns of z. z and Wx tiles live in registers; u is staged
// through LDS each iteration so every wave can read the full K=1024 of S@u.
__global__ __launch_bounds__(512)
void lista_fused_kernel(const float* __restrict__ x,
                        const float* __restrict__ Wm,
                        const float* __restrict__ Sm,
                        const float* __restrict__ threshp,
                        const int*   __restrict__ stepsp,
                        float* __restrict__ zout)
{
    extern __shared__ float smem[];   // [SDIM][ZSTRIDE]; first 512 rows reused as x-tile

    const int tid  = threadIdx.x;
    const int lane = tid & 31;
    const int wv   = tid >> 5;
    const int tile = blockIdx.x;

    const int b   = tile >> 5;          // 32 tiles per batch image (HW/NT == 32)
    const int hw0 = (tile & 31) * NT;

    // ---- stage x tile: smem[c][col] = x[b][c][hw0+col]  (contiguous 128B rows) ----
    {
        const float* xb = x + (size_t)b * FEAT * HW + hw0;
        const int col = tid & (NT - 1);
        const int c0  = tid / NT;                  // 0..15
#pragma unroll 4
        for (int c = c0; c < FEAT; c += 512 / NT)
            smem[c * ZSTRIDE + col] = xb[(size_t)c * HW + col];
    }
    __syncthreads();

    // WMMA f32 16x16x4 lane mapping (ISA 7.12.2):
    //   A: lane m = lane&15 holds row m; VGPR{0,1} hold K = ksel, ksel+1
    //   B: lane n = lane&15 holds col n; VGPR{0,1} hold K = ksel, ksel+1
    //   ksel = 0 for lanes 0-15, 2 for lanes 16-31
    const int m    = lane & 15;
    const int ksel = (lane >> 4) << 1;
    const int rhalf = (lane >> 4) << 3;            // C-tile row offset: +8 for lanes 16-31
    const int rowbase = wv * ROWS_PER_WAVE;

    // ---- Wx = W @ x_tile : 4x2 accumulator tiles per wave, kept resident ----
    v8f acc_wx[4][2];
#pragma unroll
    for (int rt = 0; rt < 4; ++rt)
#pragma unroll
        for (int ct = 0; ct < 2; ++ct) acc_wx[rt][ct] = vzero8();

#pragma unroll 1
    for (int kk = 0; kk < FEAT; kk += 4) {
        const int k0 = kk + ksel;
        v2f a[4];
#pragma unroll
        for (int rt = 0; rt < 4; ++rt)
            a[rt] = *(const v2f*)(Wm + (size_t)(rowbase + rt * 16 + m) * FEAT + k0);
        v2f bm[2];
#pragma unroll
        for (int ct = 0; ct < 2; ++ct) {
            const int n = ct * 16 + m;
            bm[ct].x = smem[(k0    ) * ZSTRIDE + n];
            bm[ct].y = smem[(k0 + 1) * ZSTRIDE + n];
        }
#pragma unroll
        for (int rt = 0; rt < 4; ++rt)
#pragma unroll
            for (int ct = 0; ct < 2; ++ct)
                acc_wx[rt][ct] = __builtin_amdgcn_wmma_f32_16x16x4_f32(
                    false, a[rt], false, bm[ct], (short)0, acc_wx[rt][ct], false, false);
    }

    const float thr    = *threshp;
    const int   nsteps = *stepsp;

    v8f acc_z[4][2];
#pragma unroll
    for (int rt = 0; rt < 4; ++rt)
#pragma unroll
        for (int ct = 0; ct < 2; ++ct) acc_z[rt][ct] = vzero8();

    for (int it = 0; it < nsteps; ++it) {
        // u = z + Wx  (registers)
#pragma unroll
        for (int rt = 0; rt < 4; ++rt)
#pragma unroll
            for (int ct = 0; ct < 2; ++ct)
                acc_z[rt][ct] = acc_z[rt][ct] + acc_wx[rt][ct];

        __syncthreads();   // previous readers of smem are done
        // stage u -> LDS so every wave sees all K=1024 rows
#pragma unroll
        for (int rt = 0; rt < 4; ++rt)
#pragma unroll
            for (int ct = 0; ct < 2; ++ct) {
                const int col = ct * 16 + m;
                const int r0  = rowbase + rt * 16 + rhalf;
#pragma unroll
                for (int j = 0; j < 8; ++j)
                    smem[(r0 + j) * ZSTRIDE + col] = acc_z[rt][ct][j];
            }
        __syncthreads();

        // sz = S @ u   (A tiles streamed from L2-resident S)
        v8f sz[4][2];
#pragma unroll
        for (int rt = 0; rt < 4; ++rt)
#pragma unroll
            for (int ct = 0; ct < 2; ++ct) sz[rt][ct] = vzero8();

#pragma unroll 1
        for (int kk = 0; kk < SDIM; kk += 4) {
            const int k0 = kk + ksel;
            v2f a[4];
#pragma unroll
            for (int rt = 0; rt < 4; ++rt)
                a[rt] = *(const v2f*)(Sm + (size_t)(rowbase + rt * 16 + m) * SDIM + k0);
            v2f bm[2];
#pragma unroll
            for (int ct = 0; ct < 2; ++ct) {
                const int n = ct * 16 + m;
                bm[ct].x = smem[(k0    ) * ZSTRIDE + n];
                bm[ct].y = smem[(k0 + 1) * ZSTRIDE + n];
            }
#pragma unroll
            for (int rt = 0; rt < 4; ++rt)
#pragma unroll
                for (int ct = 0; ct < 2; ++ct)
                    sz[rt][ct] = __builtin_amdgcn_wmma_f32_16x16x4_f32(
                        false, a[rt], false, bm[ct], (short)0, sz[rt][ct], false, false);
        }

        // z = soft_threshold(u - sz)
#pragma unroll
        for (int rt = 0; rt < 4; ++rt)
#pragma unroll
            for (int ct = 0; ct < 2; ++ct)
#pragma unroll
                for (int j = 0; j < 8; ++j) {
                    const float v   = acc_z[rt][ct][j] - sz[rt][ct][j];
                    const float mag = fabsf(v) - thr;
                    acc_z[rt][ct][j] = (mag > 0.0f) ? copysignf(mag, v) : 0.0f;
                }
    }

    // ---- stage final z through LDS, then fully coalesced [B,K,H,W] stores ----
    __syncthreads();
#pragma unroll
    for (int rt = 0; rt < 4; ++rt)
#pragma unroll
        for (int ct = 0; ct < 2; ++ct) {
            const int col = ct * 16 + m;
            const int r0  = rowbase + rt * 16 + rhalf;
#pragma unroll
            for (int j = 0; j < 8; ++j)
                smem[(r0 + j) * ZSTRIDE + col] = acc_z[rt][ct][j];
        }
    __syncthreads();
    {
        float* ob = zout + (size_t)b * SDIM * HW + hw0;
        const int col = tid & (NT - 1);
        const int r0  = tid / NT;
#pragma unroll 4
        for (int r = r0; r < SDIM; r += 512 / NT)
            ob[(size_t)r * HW + col] = smem[r * ZSTRIDE + col];
    }
}

// dict_norm = dictionary / ||row||  — trivially bandwidth-bound side output.
__global__ __launch_bounds__(256)
void dict_norm_kernel(const float* __restrict__ d, float* __restrict__ out)
{
    __shared__ float red[8];
    const int r = blockIdx.x;
    const float* row = d + (size_t)r * FEAT;

    float s = 0.0f;
    for (int c = threadIdx.x; c < FEAT; c += 256) {
        const float v = row[c];
        s += v * v;
    }
#pragma unroll
    for (int off = 16; off > 0; off >>= 1)
        s += __shfl_down(s, off, 32);
    if ((threadIdx.x & 31) == 0) red[threadIdx.x >> 5] = s;
    __syncthreads();
    if (threadIdx.x == 0) {
        float t = 0.0f;
#pragma unroll
        for (int i = 0; i < 8; ++i) t += red[i];
        red[0] = sqrtf(t);
    }
    __syncthreads();
    const float nrm = red[0];
    for (int c = threadIdx.x; c < FEAT; c += 256)
        out[(size_t)r * FEAT + c] = row[c] / nrm;
}

extern "C" void kernel_launch(void* const* d_in, const int* in_sizes, int n_in,
                              void* d_out, int out_size, void* d_ws, size_t ws_size,
                              hipStream_t stream)
{
    (void)in_sizes; (void)n_in; (void)out_size; (void)d_ws; (void)ws_size;

    const float* x     = (const float*)d_in[0];   // [32,512,32,32]
    const float* dict  = (const float*)d_in[1];   // [1024,512]
    const float* Wm    = (const float*)d_in[2];   // [1024,512]
    const float* Sm    = (const float*)d_in[3];   // [1024,1024]
    const float* thr   = (const float*)d_in[4];   // scalar f32
    const int*   steps = (const int*)d_in[5];     // scalar int

    float* zout  = (float*)d_out;                               // [32,1024,32,32]
    float* dnout = zout + (size_t)NBATCH * SDIM * HW;           // [1024,512]

    const size_t smem_bytes = (size_t)SDIM * ZSTRIDE * sizeof(float);  // 160 KB of 320 KB WGP LDS

    lista_fused_kernel<<<(NBATCH * HW) / NT, 512, smem_bytes, stream>>>(
        x, Wm, Sm, thr, steps, zout);
    dict_norm_kernel<<<SDIM, 256, 0, stream>>>(dict, dnout);
}